// Ensembler_41772851921106
// MI455X (gfx1250) — compile-verified
//
#include <hip/hip_runtime.h>
#include <stdint.h>

// Problem constants (fixed by the reference).
#define N_EXPERTS 10
#define BB 128
#define SS 4000
#define CC 5
#define NSITES (BB * SS)            // 512000 sites
#define TPB 256                     // 8 waves (wave32)
#define TILE_SITES TPB              // sites per block
#define TILE_DW (TILE_SITES * CC)   // 1280 dwords of logits per expert per block

typedef __attribute__((ext_vector_type(4))) unsigned int u32x4;
typedef __attribute__((ext_vector_type(8))) int          i32x8;
typedef __attribute__((ext_vector_type(4))) int          i32x4;

// Issue a 1-D TDM copy of `ndw` contiguous dwords from global -> LDS.
// Descriptor layout per CDNA5 ISA §8.3 (group 0) / §8.4 (group 1).
__device__ __forceinline__ void tdm_load_1d_dwords(uint32_t lds_byte_off,
                                                   const float* gsrc,
                                                   uint32_t ndw) {
  uint64_t ga = (uint64_t)(uintptr_t)gsrc;
  u32x4 g0;
  g0[0] = 1u;                                   // count=1 (valid user D#), no gather
  g0[1] = lds_byte_off;                         // lds_addr (bytes)
  g0[2] = (uint32_t)ga;                         // global_addr[31:0]
  g0[3] = (uint32_t)(ga >> 32) | (2u << 30);    // global_addr[56:32] | type=2 ("image")
  i32x8 g1;
  g1[0] = (int)(2u << 16);                      // workgroup_mask=0, data_size=2 (4B)
  g1[1] = (int)((ndw & 0xFFFFu) << 16);         // tensor_dim0[15:0] in [31:16]
  g1[2] = (int)((ndw >> 16) | (1u << 16));      // tensor_dim0[31:16] | tensor_dim1=1
  g1[3] = (int)((ndw & 0xFFFFu) << 16);         // tensor_dim1[31:16]=0 | tile_dim0=ndw
  g1[4] = 1;                                    // tile_dim1=1, tile_dim2=0
  g1[5] = (int)ndw;                             // tensor_dim0_stride[31:0]
  g1[6] = 0;                                    // stride0[47:32]=0, stride1[15:0]=0
  g1[7] = 0;                                    // stride1[47:16]=0
  i32x4 z4;
  z4[0] = 0; z4[1] = 0; z4[2] = 0; z4[3] = 0;   // groups 2/3 unused (dims <= 2)
  i32x8 z8;
  z8[0] = 0; z8[1] = 0; z8[2] = 0; z8[3] = 0;
  z8[4] = 0; z8[5] = 0; z8[6] = 0; z8[7] = 0;
  // 6-arg form on this toolchain: (g0, g1, g2, g3, extra, cpol).
  __builtin_amdgcn_tensor_load_to_lds(g0, g1, z4, z4, z8, 0);
}

__global__ __launch_bounds__(TPB) void Ensembler_41772851921106_kernel(
    const float* __restrict__ logits,   // [E, B, S, C] f32
    const float* __restrict__ noise,    // [E, B, S]    f32
    float* __restrict__ out) {          // [B, S, C]    f32
  __shared__ float tile[2][TILE_DW];    // double-buffered logits tiles (10 KB)

  const int site0 = blockIdx.x * TILE_SITES;
  const int t     = threadIdx.x;
  const int site  = site0 + t;
  const bool leader = (t < 32);         // wave 0 drives the TDM

  // Prefetch all per-expert noise values (10 independent loads in flight).
  float nz[N_EXPERTS];
#pragma unroll
  for (int e = 0; e < N_EXPERTS; ++e)
    nz[e] = noise[e * NSITES + site];

  // Raw LDS byte offsets of the two staging buffers (flat LDS addr low bits).
  const uint32_t lds0 = (uint32_t)(uintptr_t)(&tile[0][0]);
  const uint32_t lds1 = (uint32_t)(uintptr_t)(&tile[1][0]);

  if (leader)
    tdm_load_1d_dwords(lds0, logits + (uint64_t)site0 * CC, TILE_DW);

  float a0 = 0.f, a1 = 0.f, a2 = 0.f, a3 = 0.f, a4 = 0.f;

#pragma unroll
  for (int e = 0; e < N_EXPERTS; ++e) {
    if (leader) {
      if (e + 1 < N_EXPERTS) {
        // Prefetch next expert's tile, then wait until only it is outstanding
        // (TDM ops from one wave complete in order => batch e has landed).
        tdm_load_1d_dwords(((e + 1) & 1) ? lds1 : lds0,
                           logits + ((uint64_t)(e + 1) * NSITES + site0) * CC,
                           TILE_DW);
        __builtin_amdgcn_s_wait_tensorcnt(1);
      } else {
        __builtin_amdgcn_s_wait_tensorcnt(0);
      }
    }
    __syncthreads();  // publish batch e to all 8 waves

    // 5 LDS dword loads at t*5 stride: gcd(5,64)=1 -> bank-conflict-free.
    const float* p = &tile[e & 1][t * CC];
    const float v0 = p[0], v1 = p[1], v2 = p[2], v3 = p[3], v4 = p[4];

    // First-occurrence argmax (strict >), matching jnp.argmax tie-breaking.
    int k = 0; float m = v0;
    if (v1 > m) { m = v1; k = 1; }
    if (v2 > m) { m = v2; k = 2; }
    if (v3 > m) { m = v3; k = 3; }
    if (v4 > m) { m = v4; k = 4; }

    const float w = 1.0f + nz[e] * 0.001f;
    a0 += (k == 0) ? w : 0.0f;
    a1 += (k == 1) ? w : 0.0f;
    a2 += (k == 2) ? w : 0.0f;
    a3 += (k == 3) ? w : 0.0f;
    a4 += (k == 4) ? w : 0.0f;

    __syncthreads();  // all reads of buf[e&1] done before it is re-filled (e+2)
  }

  float* o = out + (uint64_t)site * CC;
  o[0] = a0; o[1] = a1; o[2] = a2; o[3] = a3; o[4] = a4;
}

extern "C" void kernel_launch(void* const* d_in, const int* in_sizes, int n_in,
                              void* d_out, int out_size, void* d_ws, size_t ws_size,
                              hipStream_t stream) {
  const float* logits = (const float*)d_in[0];   // [10,128,4000,5] f32
  const float* noise  = (const float*)d_in[1];   // [10,128,4000]   f32
  float* out = (float*)d_out;                    // [128,4000,5]    f32
  (void)in_sizes; (void)n_in; (void)out_size; (void)d_ws; (void)ws_size;

  dim3 grid(NSITES / TILE_SITES);  // 2000 blocks, exact cover (no tail)
  Ensembler_41772851921106_kernel<<<grid, dim3(TPB), 0, stream>>>(logits, noise, out);
}